// ConvGen220_65481071408809
// MI455X (gfx1250) — compile-verified
//
#include <hip/hip_runtime.h>

typedef __attribute__((ext_vector_type(2))) float v2f;
typedef __attribute__((ext_vector_type(8))) float v8f;

#define BATCH 1024
#define IH 224
#define IW 224
#define OH 223
#define OW 223
#define FLAT (OH * OW)   // 49729
#define AD 4

// ---------------------------------------------------------------------------
// Kernel 1: fused conv2x2(+bias) -> sigmoid -> embed-GEMM reduction.
// Memory-bound: streams x (205 MB) exactly once; embed_w served from L2.
// Each block owns NB batches so embed_w L2 traffic is divided by NB.
// ---------------------------------------------------------------------------
#define NB 4
#define K1_THREADS 256

__global__ void __launch_bounds__(K1_THREADS)
fused_embed_kernel(const float* __restrict__ x,
                   const float* __restrict__ conv_w,
                   const float* __restrict__ conv_b,
                   const float* __restrict__ embed_w,
                   const float* __restrict__ embed_b,
                   float* __restrict__ E) {
    const int tid = threadIdx.x;
    const int b0  = blockIdx.x * NB;

    const float w00 = conv_w[0], w01 = conv_w[1];
    const float w10 = conv_w[2], w11 = conv_w[3];
    const float cb  = conv_b[0];

    float acc[NB][AD];
#pragma unroll
    for (int b = 0; b < NB; ++b)
#pragma unroll
        for (int d = 0; d < AD; ++d) acc[b][d] = 0.0f;

    for (int p = tid; p < FLAT; p += K1_THREADS) {
        const int i    = p / OW;
        const int j    = p - i * OW;
        const int xoff = i * IW + j;

        float wv[AD];
#pragma unroll
        for (int d = 0; d < AD; ++d) wv[d] = embed_w[(size_t)d * FLAT + p];

#pragma unroll
        for (int b = 0; b < NB; ++b) {
            const float* xb = x + (size_t)(b0 + b) * (IH * IW) + xoff;
            float c = fmaf(xb[0], w00,
                      fmaf(xb[1], w01,
                      fmaf(xb[IW], w10,
                      fmaf(xb[IW + 1], w11, cb))));
            // sigmoid(c): rcp(1 + exp(-c)) -> v_exp_f32 + v_rcp_f32
            float s = __builtin_amdgcn_rcpf(1.0f + __expf(-c));
#pragma unroll
            for (int d = 0; d < AD; ++d) acc[b][d] = fmaf(s, wv[d], acc[b][d]);
        }
    }

    // Block-wide tree reduction of NB*AD = 16 partial sums.
    __shared__ float red[NB * AD][K1_THREADS];   // 16 KB
#pragma unroll
    for (int v = 0; v < NB * AD; ++v) red[v][tid] = acc[v / AD][v % AD];
    __syncthreads();
    for (int s = K1_THREADS / 2; s > 0; s >>= 1) {
        if (tid < s) {
#pragma unroll
            for (int v = 0; v < NB * AD; ++v) red[v][tid] += red[v][tid + s];
        }
        __syncthreads();
    }
    if (tid < NB * AD) {
        const int b = tid / AD, d = tid % AD;
        E[(size_t)(b0 + b) * AD + d] = red[tid][0] + embed_b[d];
    }
}

// ---------------------------------------------------------------------------
// Kernel 2: q = E @ rotation, k = E @ entangle   (tiny: 1024 x 4 x 4)
// ---------------------------------------------------------------------------
__global__ void qk_kernel(const float* __restrict__ E,
                          const float* __restrict__ rot,
                          const float* __restrict__ ent,
                          float* __restrict__ q,
                          float* __restrict__ k) {
    const int b = blockIdx.x * blockDim.x + threadIdx.x;
    if (b >= BATCH) return;
    float e[AD];
#pragma unroll
    for (int d = 0; d < AD; ++d) e[d] = E[b * AD + d];
#pragma unroll
    for (int d = 0; d < AD; ++d) {
        float qa = 0.0f, ka = 0.0f;
#pragma unroll
        for (int j = 0; j < AD; ++j) {
            qa = fmaf(e[j], rot[j * AD + d], qa);
            ka = fmaf(e[j], ent[j * AD + d], ka);
        }
        q[b * AD + d] = qa;
        k[b * AD + d] = ka;
    }
}

// ---------------------------------------------------------------------------
// Kernel 3: scores = (q @ k^T) * rsqrt(4) via V_WMMA_F32_16X16X4_F32.
// K = ATTN_DIM = 4 matches the WMMA K exactly; full f32, no conversion.
// One wave per 16x16 output tile; EXEC is all-ones (32 threads, no branches).
// A 16x4 layout: lane l -> row m=l%16, VGPR pair holds K = 2*(l/16), +1.
// B 4x16 layout: lane l -> col n=l%16, same K striping. B[kd][n] = k[n][kd].
// D 16x16 layout: VGPR i, lane l -> (m = i + 8*(l/16), n = l%16).
// ---------------------------------------------------------------------------
__global__ void __launch_bounds__(32)
scores_kernel(const float* __restrict__ q,
              const float* __restrict__ k,
              float* __restrict__ S) {
    const int lane = threadIdx.x;          // 0..31
    const int n0   = blockIdx.x * 16;      // key tile
    const int m0   = blockIdx.y * 16;      // query tile
    const int r    = lane & 15;
    const int h    = lane >> 4;            // K-half select

    v2f a, bmat;
    a.x    = q[(size_t)(m0 + r) * AD + 2 * h + 0];
    a.y    = q[(size_t)(m0 + r) * AD + 2 * h + 1];
    bmat.x = k[(size_t)(n0 + r) * AD + 2 * h + 0];
    bmat.y = k[(size_t)(n0 + r) * AD + 2 * h + 1];

    v8f c = {};
    // 8 args: (neg_a, A, neg_b, B, c_mod, C, reuse_a, reuse_b)
    c = __builtin_amdgcn_wmma_f32_16x16x4_f32(
        /*neg_a=*/false, a, /*neg_b=*/false, bmat,
        /*c_mod=*/(short)0, c, /*reuse_a=*/false, /*reuse_b=*/false);

#pragma unroll
    for (int i = 0; i < 8; ++i) {
        const int m = i + 8 * h;
        S[(size_t)(m0 + m) * BATCH + (n0 + r)] = c[i] * 0.5f;  // 1/sqrt(4)
    }
}

// ---------------------------------------------------------------------------
// Kernel 4: row softmax over S, attn = P @ E, mean, sigmoid, output.
// One block per query row; S row (4 KB) hits L2 on the second pass.
// ---------------------------------------------------------------------------
#define K4_THREADS 256

__global__ void __launch_bounds__(K4_THREADS)
softmax_attn_kernel(const float* __restrict__ S,
                    const float* __restrict__ E,
                    float* __restrict__ out) {
    const int b   = blockIdx.x;
    const int tid = threadIdx.x;
    const float* row = S + (size_t)b * BATCH;

    __shared__ float sred[5][K4_THREADS];

    // Pass 1: row max.
    float m = -3.402823466e38f;
    for (int j = tid; j < BATCH; j += K4_THREADS) m = fmaxf(m, row[j]);
    sred[0][tid] = m;
    __syncthreads();
    for (int s = K4_THREADS / 2; s > 0; s >>= 1) {
        if (tid < s) sred[0][tid] = fmaxf(sred[0][tid], sred[0][tid + s]);
        __syncthreads();
    }
    m = sred[0][0];
    __syncthreads();

    // Pass 2: exp-sum and weighted accumulation of E (fused P @ E).
    float sum = 0.0f;
    float acc[AD] = {0.0f, 0.0f, 0.0f, 0.0f};
    for (int j = tid; j < BATCH; j += K4_THREADS) {
        const float p = __expf(row[j] - m);
        sum += p;
#pragma unroll
        for (int d = 0; d < AD; ++d) acc[d] = fmaf(p, E[(size_t)j * AD + d], acc[d]);
    }
    sred[0][tid] = sum;
#pragma unroll
    for (int d = 0; d < AD; ++d) sred[1 + d][tid] = acc[d];
    __syncthreads();
    for (int s = K4_THREADS / 2; s > 0; s >>= 1) {
        if (tid < s) {
#pragma unroll
            for (int v = 0; v < 5; ++v) sred[v][tid] += sred[v][tid + s];
        }
        __syncthreads();
    }
    if (tid == 0) {
        const float inv  = 1.0f / sred[0][0];
        const float mean = (sred[1][0] + sred[2][0] + sred[3][0] + sred[4][0]) * 0.25f * inv;
        const float prob = __builtin_amdgcn_rcpf(1.0f + __expf(-mean));
        out[(size_t)b * 2 + 0] = prob;
        out[(size_t)b * 2 + 1] = 1.0f - prob;
    }
}

// ---------------------------------------------------------------------------
// kernel_launch
// Inputs (setup_inputs order): x, conv_w, conv_b, embed_w, embed_b,
//                              rotation, entangle  (all float32)
// Output: [1024, 2] float32.
// Workspace layout (floats): E @0 (4K), q @4096 (4K), k @8192 (4K),
//                            S @12288 (1M floats = 4 MB).
// ---------------------------------------------------------------------------
extern "C" void kernel_launch(void* const* d_in, const int* in_sizes, int n_in,
                              void* d_out, int out_size, void* d_ws, size_t ws_size,
                              hipStream_t stream) {
    const float* x       = (const float*)d_in[0];
    const float* conv_w  = (const float*)d_in[1];
    const float* conv_b  = (const float*)d_in[2];
    const float* embed_w = (const float*)d_in[3];
    const float* embed_b = (const float*)d_in[4];
    const float* rot     = (const float*)d_in[5];
    const float* ent     = (const float*)d_in[6];
    float* out = (float*)d_out;

    float* ws = (float*)d_ws;
    float* E = ws;                 // 1024 * 4
    float* q = ws + 4096;          // 1024 * 4
    float* k = ws + 8192;          // 1024 * 4
    float* S = ws + 12288;         // 1024 * 1024

    fused_embed_kernel<<<BATCH / NB, K1_THREADS, 0, stream>>>(
        x, conv_w, conv_b, embed_w, embed_b, E);
    qk_kernel<<<BATCH / 256, 256, 0, stream>>>(E, rot, ent, q, k);
    scores_kernel<<<dim3(BATCH / 16, BATCH / 16), 32, 0, stream>>>(q, k, S);
    softmax_attn_kernel<<<BATCH, K4_THREADS, 0, stream>>>(S, E, out);
}